// TorchTemplateClassifier_14431090114572
// MI455X (gfx1250) — compile-verified
//
#include <hip/hip_runtime.h>
#include <hip/hip_bf16.h>

#define B_      32
#define S_      4096
#define H_      1024
#define NSPANS_ 16
#define NCLS_   25

typedef float v2f __attribute__((ext_vector_type(2)));
typedef float v8f __attribute__((ext_vector_type(8)));

// ---------------------------------------------------------------------------
// Kernel 0: zero the pooled accumulator (d_ws is poisoned; must self-init).
// ---------------------------------------------------------------------------
__global__ void zero_f32(float* __restrict__ p, int n) {
    int i = blockIdx.x * blockDim.x + threadIdx.x;
    if (i < n) p[i] = 0.0f;
}

// ---------------------------------------------------------------------------
// Kernel 1: span-sum accumulation.
// grid = (H/256, NSPANS, B), block = 256.
// Block (x=dchunk, y=span n, z=batch b): if span n is valid under the
// "break at first (0,0)" rule, stream its tokens (coalesced 1KB rows) and
// atomicAdd into pooled[b][d]. Overlapping spans double-count, as reference.
// ---------------------------------------------------------------------------
__global__ void span_accum(const float* __restrict__ emb,
                           const int*   __restrict__ spans,
                           float*       __restrict__ pooled) {
    const int b = blockIdx.z;
    const int n = blockIdx.y;
    const int d = blockIdx.x * 256 + threadIdx.x;

    const int* sp = spans + b * NSPANS_ * 2;
    int s0 = 0, s1 = 0;
    for (int j = 0; j <= n; ++j) {           // break semantics: all earlier
        s0 = sp[2 * j];                      // spans must be non-(0,0)
        s1 = sp[2 * j + 1];
        if (s0 == 0 && s1 == 0) return;
    }

    float acc = 0.0f;
    const float* base = emb + (size_t)b * S_ * H_ + d;
    for (int t = s0; t < s1; ++t)
        acc += base[(size_t)t * H_];
    atomicAdd(&pooled[b * H_ + d], acc);
}

// ---------------------------------------------------------------------------
// Kernel 2: divide by total token count. grid = (H/256, B), block = 256.
// ---------------------------------------------------------------------------
__global__ void span_norm(const int* __restrict__ spans,
                          float*     __restrict__ pooled) {
    const int b = blockIdx.y;
    const int d = blockIdx.x * 256 + threadIdx.x;
    const int* sp = spans + b * NSPANS_ * 2;
    int total = 0;
    for (int j = 0; j < NSPANS_; ++j) {
        int s0 = sp[2 * j], s1 = sp[2 * j + 1];
        if (s0 == 0 && s1 == 0) break;
        total += (s1 - s0);
    }
    float inv = (total > 0) ? (1.0f / (float)total) : 0.0f;
    pooled[b * H_ + d] *= inv;
}

// ---------------------------------------------------------------------------
// Kernel 3: fc1 + bias + relu via V_WMMA_F32_16X16X4_F32 (full fp32 path).
// grid = (H/16 = 64 N-tiles, B/16 = 2 M-tiles), block = 32 (one wave32,
// EXEC all-ones as WMMA requires).
// A layout (16x4 f32): lanes 0-15 -> M=lane, VGPR0/1 = K0/K1;
//                      lanes 16-31 -> M=lane-16, VGPR0/1 = K2/K3.
// B layout (4x16 f32): lanes hold N; VGPR0/1 = (K0,K1) lanes 0-15,
//                      (K2,K3) lanes 16-31.
// C/D (16x16 f32): VGPR r -> M = r + 8*(lane>=16), N = lane&15.
// ---------------------------------------------------------------------------
__global__ void fc1_wmma(const float* __restrict__ pooled,
                         const float* __restrict__ W1,
                         const float* __restrict__ b1,
                         float*       __restrict__ h) {
    const int lane  = threadIdx.x;     // 0..31
    const int ntile = blockIdx.x;      // 0..63
    const int mtile = blockIdx.y;      // 0..1
    const int half  = lane >> 4;       // 0 or 1
    const int lc    = lane & 15;
    const int row   = mtile * 16 + lc; // A row handled by this lane
    const int col   = ntile * 16 + lc; // B/D column handled by this lane

    v8f c = {};
    for (int k0 = 0; k0 < H_; k0 += 4) {
        const int ka = k0 + half * 2;
        v2f a, bm;
        a.x  = pooled[row * H_ + ka];
        a.y  = pooled[row * H_ + ka + 1];
        bm.x = W1[(size_t)ka * H_ + col];
        bm.y = W1[(size_t)(ka + 1) * H_ + col];
        // 8 args: (neg_a, A, neg_b, B, c_mod, C, reuse_a, reuse_b)
        c = __builtin_amdgcn_wmma_f32_16x16x4_f32(
                false, a, false, bm, (short)0, c, false, false);
    }

    const float bias = b1[col];
#pragma unroll
    for (int r = 0; r < 8; ++r) {
        const int m = r + half * 8;
        float v = c[r] + bias;
        h[(mtile * 16 + m) * H_ + col] = v > 0.0f ? v : 0.0f;
    }
}

// ---------------------------------------------------------------------------
// Kernel 4: fc2 + bias + sigmoid. grid = B, block = 32 (25 active lanes).
// ---------------------------------------------------------------------------
__global__ void fc2_sigmoid(const float* __restrict__ h,
                            const float* __restrict__ W2,
                            const float* __restrict__ b2,
                            float*       __restrict__ out) {
    const int b = blockIdx.x;
    const int cidx = threadIdx.x;
    if (cidx >= NCLS_) return;
    float acc = b2[cidx];
    const float* hr = h + b * H_;
    for (int k = 0; k < H_; ++k)
        acc += hr[k] * W2[k * NCLS_ + cidx];
    out[b * NCLS_ + cidx] = 1.0f / (1.0f + __expf(-acc));
}

// ---------------------------------------------------------------------------
extern "C" void kernel_launch(void* const* d_in, const int* in_sizes, int n_in,
                              void* d_out, int out_size, void* d_ws, size_t ws_size,
                              hipStream_t stream) {
    const float* emb   = (const float*)d_in[0];   // [B,S,H] f32
    const int*   spans = (const int*)  d_in[1];   // [B,16,2] i32
    const float* W1    = (const float*)d_in[2];   // [H,H]
    const float* b1    = (const float*)d_in[3];   // [H]
    const float* W2    = (const float*)d_in[4];   // [H,25]
    const float* b2    = (const float*)d_in[5];   // [25]
    float*       out   = (float*)d_out;           // [B,25]

    float* pooled = (float*)d_ws;                 // [B,H]  (128 KB)
    float* hbuf   = pooled + B_ * H_;             // [B,H]  (128 KB)

    zero_f32<<<(B_ * H_ + 255) / 256, 256, 0, stream>>>(pooled, B_ * H_);
    span_accum<<<dim3(H_ / 256, NSPANS_, B_), 256, 0, stream>>>(emb, spans, pooled);
    span_norm<<<dim3(H_ / 256, B_), 256, 0, stream>>>(spans, pooled);
    fc1_wmma<<<dim3(H_ / 16, B_ / 16), 32, 0, stream>>>(pooled, W1, b1, hbuf);
    fc2_sigmoid<<<B_, 32, 0, stream>>>(hbuf, W2, b2, out);
}